// CausalSelfAttention_86603720556556
// MI455X (gfx1250) — compile-verified
//
#include <hip/hip_runtime.h>

typedef __attribute__((ext_vector_type(2))) float v2f;
typedef __attribute__((ext_vector_type(8))) float v8f;
typedef __attribute__((ext_vector_type(4))) unsigned int v4u;
typedef __attribute__((ext_vector_type(4))) int v4i;
typedef __attribute__((ext_vector_type(8))) int v8i;

#define BB 4
#define TT 2048
#define CC 1024
#define NHH 16
#define HDD 64

static __device__ __forceinline__ v8f wmma_f32(v2f a, v2f b, v8f c) {
  // D = A(16x4,f32) * B(4x16,f32) + C(16x16,f32)
  return __builtin_amdgcn_wmma_f32_16x16x4_f32(false, a, false, b, (short)0, c,
                                               false, false);
}

static __device__ __forceinline__ unsigned int lds_off(const void* p) {
  // generic -> LDS address space: flat LDS aperture keeps offset in addr[31:0]
  return (unsigned int)(uintptr_t)(__attribute__((address_space(3))) const void*)p;
}

// ---------------------------------------------------------------------------
// Tensor Data Mover: async 2D tile load (4-byte elements) global -> LDS.
// tile0 = elems per row, tile1 = rows, stride0 = row stride in elems.
// LDS rows padded: pad_iv encodes 2^(iv+1) dwords between pads, pad_amt+1
// dwords of padding (ISA cdna5 sec 8.4).
// ---------------------------------------------------------------------------
static __device__ __forceinline__ void tdm_load_2d(
    unsigned int lds, const void* gptr, unsigned int tile0, unsigned int tile1,
    unsigned long long stride0, unsigned int pad_iv, unsigned int pad_amt) {
  unsigned long long ga = (unsigned long long)(uintptr_t)gptr;
  v4u g0;
  g0.x = 1u;                // count=1, user mode, no gather
  g0.y = lds;               // LDS byte address
  g0.z = (unsigned int)ga;  // global addr [31:0]
  g0.w = (unsigned int)((ga >> 32) & 0x01FFFFFFu) | (2u << 30);  // addr|type=2
  const unsigned int dim0 = 0x40000000u, dim1 = 0x40000000u;  // no OOB clamp
  v8i g1;
  g1[0] = (int)((2u << 16) | (1u << 20) | (pad_iv << 22) | (pad_amt << 25));
  g1[1] = (int)((dim0 & 0xFFFFu) << 16);
  g1[2] = (int)((dim0 >> 16) | ((dim1 & 0xFFFFu) << 16));
  g1[3] = (int)((dim1 >> 16) | (tile0 << 16));
  g1[4] = (int)tile1;  // tile_dim1; tile_dim2 = 0
  g1[5] = (int)(unsigned int)stride0;
  g1[6] = (int)(unsigned int)(stride0 >> 32);  // stride1 = 0 (2D)
  g1[7] = 0;
  v4i g2 = {0, 0, 0, 0};
  v4i g3 = {0, 0, 0, 0};
  v8i g4 = {0, 0, 0, 0, 0, 0, 0, 0};  // unused for 2D tiles
  __builtin_amdgcn_tensor_load_to_lds(g0, g1, g2, g3, g4, 0);
}

// ---------------------------------------------------------------------------
// Kernel 1: qkv = x @ W_qkv + b_qkv, RoPE on q/k, scatter to [B,NH,T,HD]
// block tile 64(M) x 128(N), K-step 16; TDM double-buffered staging.
// ---------------------------------------------------------------------------
__global__ __launch_bounds__(256) void qkv_rope_kernel(
    const float* __restrict__ x, const float* __restrict__ Wqkv,
    const float* __restrict__ bqkv, float* __restrict__ qws,
    float* __restrict__ kws, float* __restrict__ vws) {
  __shared__ __align__(16) float lA[2][64 * 18];    // [m][k], stride 18
  __shared__ __align__(16) float lB[2][16 * 130];   // [k][n], stride 130

  const int tid = threadIdx.x;
  const int m0 = blockIdx.y * 64;
  const int n0 = blockIdx.x * 128;
  const int w = tid >> 5, lane = tid & 31;
  const int hl = lane >> 4, ln = lane & 15;
  const int wm = w & 3, wn = w >> 2;

  v8f acc[4] = {};
  const int NK = CC / 16;

  if (w == 0) {
    tdm_load_2d(lds_off(&lA[0][0]), x + (size_t)m0 * CC, 16, 64, CC, 3, 1);
    tdm_load_2d(lds_off(&lB[0][0]), Wqkv + n0, 128, 16, 3 * CC, 6, 1);
  }

  for (int i = 0; i < NK; ++i) {
    const int cur = i & 1;
    if (w == 0) __builtin_amdgcn_s_wait_tensorcnt(0);
    __syncthreads();
    if (w == 0 && i + 1 < NK) {
      const int k0n = (i + 1) * 16;
      tdm_load_2d(lds_off(&lA[cur ^ 1][0]), x + (size_t)m0 * CC + k0n, 16, 64,
                  CC, 3, 1);
      tdm_load_2d(lds_off(&lB[cur ^ 1][0]),
                  Wqkv + (size_t)k0n * (3 * CC) + n0, 128, 16, 3 * CC, 6, 1);
    }
    const float* Ac = &lA[cur][0];
    const float* Bc = &lB[cur][0];
#pragma unroll
    for (int cc = 0; cc < 4; ++cc) {
      v2f a = *(const v2f*)&Ac[(wm * 16 + ln) * 18 + cc * 4 + 2 * hl];
      const int r0 = (cc * 4 + 2 * hl) * 130;
#pragma unroll
      for (int ns = 0; ns < 4; ++ns) {
        const int nn = wn * 64 + ns * 16 + ln;
        v2f b = {Bc[r0 + nn], Bc[r0 + 130 + nn]};
        acc[ns] = wmma_f32(a, b, acc[ns]);
      }
    }
  }

  // epilogue: bias, RoPE (for q/k chunks), scatter to [B,NH,T,HD]
  const int chunk = (n0 + wn * 64) >> 6;  // 0..47 (16 q, 16 k, 16 v heads)
  const int mat = chunk >> 4;
  const int h = chunk & 15;
  float* outp = (mat == 0) ? qws : (mat == 1) ? kws : vws;

#pragma unroll
  for (int ns = 0; ns < 4; ++ns) {
    float bias = bqkv[n0 + wn * 64 + ns * 16 + ln];
#pragma unroll
    for (int r = 0; r < 8; ++r) acc[ns][r] += bias;
  }

  float res[4][8];
  if (mat < 2) {
    float cs[2][8], sn[2][8];
#pragma unroll
    for (int par = 0; par < 2; ++par) {
      float f = (float)(par * 16 + ln);  // freq index d % 32
      float invf = __expf(-(2.0f * f / (float)HDD) * 9.210340371976184f);
#pragma unroll
      for (int r = 0; r < 8; ++r) {
        int row = m0 + wm * 16 + r + 8 * hl;
        float t = (float)(row & (TT - 1));
        float ang = t * invf;
        cs[par][r] = cosf(ang);
        sn[par][r] = sinf(ang);
      }
    }
#pragma unroll
    for (int ns = 0; ns < 4; ++ns) {
      float sgn = (ns < 2) ? -1.0f : 1.0f;  // rotate_half sign
      int par = ns & 1;
#pragma unroll
      for (int r = 0; r < 8; ++r)
        res[ns][r] =
            acc[ns][r] * cs[par][r] + sgn * acc[(ns + 2) & 3][r] * sn[par][r];
    }
  } else {
#pragma unroll
    for (int ns = 0; ns < 4; ++ns)
#pragma unroll
      for (int r = 0; r < 8; ++r) res[ns][r] = acc[ns][r];
  }

#pragma unroll
  for (int ns = 0; ns < 4; ++ns)
#pragma unroll
    for (int r = 0; r < 8; ++r) {
      int row = m0 + wm * 16 + r + 8 * hl;
      int bidx = row >> 11, t = row & (TT - 1);
      outp[(size_t)((bidx * NHH + h) * TT + t) * HDD + ns * 16 + ln] =
          res[ns][r];
    }
}

// ---------------------------------------------------------------------------
// Kernel 2: flash attention per (b,h); block = 128 q rows, wave = 16 q rows
// K/V tiles staged via TDM, double-buffered.
// ---------------------------------------------------------------------------
__global__ __launch_bounds__(256) void attn_kernel(
    const float* __restrict__ qws, const float* __restrict__ kws,
    const float* __restrict__ vws, float* __restrict__ yws) {
  __shared__ __align__(16) float lK[2][16 * 66];   // [key][dim], stride 66
  __shared__ __align__(16) float lV[2][16 * 66];   // [key][dim], stride 66
  __shared__ __align__(16) float lP[8 * 16 * 18];  // per-wave P staging

  const int tid = threadIdx.x;
  const int w = tid >> 5, lane = tid & 31;
  const int hl = lane >> 4, ln = lane & 15;

  const int qt = blockIdx.x & 15;  // T/128 = 16 q tiles
  const int bh = blockIdx.x >> 4;  // 0..63
  const size_t base = (size_t)bh * TT * HDD;

  const int q0 = qt * 128;
  const int qw = q0 + 16 * w;       // wave's first q row
  const int myKT = qw >> 4;         // last key tile (inclusive) for this wave
  const int ktEnd = (q0 >> 4) + 8;  // exclusive, whole block

  // preload Q as A-fragments, pre-scaled by 1/sqrt(HD)
  v2f qa[16];
  {
    const float* qp = qws + base + (size_t)(qw + ln) * HDD;
#pragma unroll
    for (int c = 0; c < 16; ++c) {
      v2f t = *(const v2f*)&qp[c * 4 + 2 * hl];
      qa[c] = t * 0.125f;
    }
  }

  v8f accO[4] = {};
  float mrow[8], lrow[8];
#pragma unroll
  for (int r = 0; r < 8; ++r) {
    mrow[r] = -3.0e38f;
    lrow[r] = 0.0f;
  }

  const float* kbase = kws + base;
  const float* vbase = vws + base;
  if (w == 0) {
    tdm_load_2d(lds_off(&lK[0][0]), kbase, 64, 16, HDD, 5, 1);
    tdm_load_2d(lds_off(&lV[0][0]), vbase, 64, 16, HDD, 5, 1);
  }

  for (int kt = 0; kt < ktEnd; ++kt) {
    const int cur = kt & 1;
    if (w == 0) __builtin_amdgcn_s_wait_tensorcnt(0);
    __syncthreads();
    if (w == 0 && kt + 1 < ktEnd) {
      const size_t off = (size_t)(kt + 1) * 16 * HDD;
      tdm_load_2d(lds_off(&lK[cur ^ 1][0]), kbase + off, 64, 16, HDD, 5, 1);
      tdm_load_2d(lds_off(&lV[cur ^ 1][0]), vbase + off, 64, 16, HDD, 5, 1);
    }
    if (kt > myKT) continue;  // wave-uniform: EXEC stays full for WMMA

    const float* lKc = &lK[cur][0];
    const float* lVc = &lV[cur][0];

    // S = (Q/sqrt(HD)) @ K^T : 16 WMMAs over HD=64
    v8f s = {};
#pragma unroll
    for (int c = 0; c < 16; ++c) {
      v2f b = *(const v2f*)&lKc[ln * 66 + c * 4 + 2 * hl];
      s = wmma_f32(qa[c], b, s);
    }

    if (kt == myKT) {  // causal mask on diagonal tile
      int kg = kt * 16 + ln;
#pragma unroll
      for (int r = 0; r < 8; ++r) {
        int qr = qw + r + 8 * hl;
        if (kg > qr) s[r] = -3.0e38f;
      }
    }

    // online softmax (row reductions within 16-lane halves)
    float p[8];
#pragma unroll
    for (int r = 0; r < 8; ++r) {
      float sm = s[r];
#pragma unroll
      for (int off = 8; off >= 1; off >>= 1)
        sm = fmaxf(sm, __shfl_xor(sm, off, 32));
      float mn = fmaxf(mrow[r], sm);
      float alpha = __expf(mrow[r] - mn);
      mrow[r] = mn;
      float pv = __expf(s[r] - mn);
      p[r] = pv;
      float rs = pv;
#pragma unroll
      for (int off = 8; off >= 1; off >>= 1) rs += __shfl_xor(rs, off, 32);
      lrow[r] = lrow[r] * alpha + rs;
#pragma unroll
      for (int nt = 0; nt < 4; ++nt) accO[nt][r] *= alpha;
    }

    // stage P (C-layout -> A-layout) through per-wave LDS
    float* myP = &lP[w * 288];
#pragma unroll
    for (int r = 0; r < 8; ++r) myP[(r + 8 * hl) * 18 + ln] = p[r];
    asm volatile("s_wait_dscnt 0" ::: "memory");  // cross-lane LDS dep

    // O += P @ V : 16 WMMAs
#pragma unroll
    for (int cc = 0; cc < 4; ++cc) {
      v2f a = *(const v2f*)&myP[ln * 18 + cc * 4 + 2 * hl];
      const int r0 = (cc * 4 + 2 * hl) * 66;
#pragma unroll
      for (int nt = 0; nt < 4; ++nt) {
        v2f b = {lVc[r0 + nt * 16 + ln], lVc[r0 + 66 + nt * 16 + ln]};
        accO[nt] = wmma_f32(a, b, accO[nt]);
      }
    }
  }

  // normalize and store y in [B,NH,T,HD]
#pragma unroll
  for (int r = 0; r < 8; ++r) {
    float inv = 1.0f / lrow[r];
    int qr = qw + r + 8 * hl;
#pragma unroll
    for (int nt = 0; nt < 4; ++nt)
      yws[base + (size_t)qr * HDD + nt * 16 + ln] = accO[nt][r] * inv;
  }
}

// ---------------------------------------------------------------------------
// Kernel 3: out = y @ W_proj + b_proj  (A gathered from head-major y)
// ---------------------------------------------------------------------------
__global__ __launch_bounds__(256) void proj_kernel(
    const float* __restrict__ yws, const float* __restrict__ Wp,
    const float* __restrict__ bp, float* __restrict__ out) {
  __shared__ __align__(16) float lA[2][64 * 18];   // [m][k], stride 18
  __shared__ __align__(16) float lB[2][16 * 130];  // [k][n], stride 130

  const int tid = threadIdx.x;
  const int m0 = blockIdx.y * 64;
  const int n0 = blockIdx.x * 128;
  const int w = tid >> 5, lane = tid & 31;
  const int hl = lane >> 4, ln = lane & 15;
  const int wm = w & 3, wn = w >> 2;

  const int bidx = m0 >> 11, t0 = m0 & (TT - 1);

  v8f acc[4] = {};
  const int NK = CC / 16;

  if (w == 0) {
    tdm_load_2d(lds_off(&lA[0][0]),
                yws + (size_t)(bidx * NHH + 0) * TT * HDD + (size_t)t0 * HDD,
                16, 64, HDD, 3, 1);
    tdm_load_2d(lds_off(&lB[0][0]), Wp + n0, 128, 16, CC, 6, 1);
  }

  for (int i = 0; i < NK; ++i) {
    const int cur = i & 1;
    if (w == 0) __builtin_amdgcn_s_wait_tensorcnt(0);
    __syncthreads();
    if (w == 0 && i + 1 < NK) {
      const int k0n = (i + 1) * 16;
      const int h = k0n >> 6, d0 = k0n & 63;
      tdm_load_2d(lds_off(&lA[cur ^ 1][0]),
                  yws + (size_t)(bidx * NHH + h) * TT * HDD +
                      (size_t)t0 * HDD + d0,
                  16, 64, HDD, 3, 1);
      tdm_load_2d(lds_off(&lB[cur ^ 1][0]), Wp + (size_t)k0n * CC + n0, 128,
                  16, CC, 6, 1);
    }
    const float* Ac = &lA[cur][0];
    const float* Bc = &lB[cur][0];
#pragma unroll
    for (int cc = 0; cc < 4; ++cc) {
      v2f a = *(const v2f*)&Ac[(wm * 16 + ln) * 18 + cc * 4 + 2 * hl];
      const int r0 = (cc * 4 + 2 * hl) * 130;
#pragma unroll
      for (int ns = 0; ns < 4; ++ns) {
        const int nn = wn * 64 + ns * 16 + ln;
        v2f b = {Bc[r0 + nn], Bc[r0 + 130 + nn]};
        acc[ns] = wmma_f32(a, b, acc[ns]);
      }
    }
  }

#pragma unroll
  for (int ns = 0; ns < 4; ++ns) {
    float bias = bp[n0 + wn * 64 + ns * 16 + ln];
#pragma unroll
    for (int r = 0; r < 8; ++r) {
      int row = m0 + wm * 16 + r + 8 * hl;
      out[(size_t)row * CC + n0 + wn * 64 + ns * 16 + ln] = acc[ns][r] + bias;
    }
  }
}

// ---------------------------------------------------------------------------
extern "C" void kernel_launch(void* const* d_in, const int* in_sizes, int n_in,
                              void* d_out, int out_size, void* d_ws,
                              size_t ws_size, hipStream_t stream) {
  (void)in_sizes; (void)n_in; (void)out_size; (void)ws_size;
  const float* x = (const float*)d_in[0];
  const float* Wqkv = (const float*)d_in[1];
  const float* bqkv = (const float*)d_in[2];
  const float* Wp = (const float*)d_in[3];
  const float* bp = (const float*)d_in[4];
  float* out = (float*)d_out;

  const size_t elems = (size_t)BB * NHH * TT * HDD;  // 8M floats each
  float* qws = (float*)d_ws;
  float* kws = qws + elems;
  float* vws = kws + elems;
  float* yws = vws + elems;  // total 128 MB of workspace

  qkv_rope_kernel<<<dim3(3 * CC / 128, BB * TT / 64), 256, 0, stream>>>(
      x, Wqkv, bqkv, qws, kws, vws);
  attn_kernel<<<dim3(BB * NHH * (TT / 128)), 256, 0, stream>>>(qws, kws, vws,
                                                               yws);
  proj_kernel<<<dim3(CC / 128, BB * TT / 64), 256, 0, stream>>>(yws, Wp, bp,
                                                                out);
}